// WinogradConv2d_67808943669459
// MI455X (gfx1250) — compile-verified
//
#include <hip/hip_runtime.h>

typedef float v2f __attribute__((ext_vector_type(2)));
typedef float v8f __attribute__((ext_vector_type(8)));

#define H_DIM 56
#define W_DIM 56
#define TH 28
#define TW 28
#define NB 16
#define CIN 256
#define OCH 256
#define TILES_TOTAL (NB * TH * TW)   // 12544
#define TILES_PER_N (TH * TW)        // 784
#define TPB 16                       // tiles per workgroup (GEMM N-block)
#define OPB 64                       // output channels per workgroup (GEMM M-block)
#define OFRAGS (OPB / 16)            // 4 A-fragments per k-step
#define CCHUNK 32                    // K chunk staged in LDS per iteration
#define VSTRIDE 34                   // padded c-stride in LDS (8B aligned, conflict padding)

// ---------------------------------------------------------------------------
// Kernel 1: weight transform  U[il][o][c] = (G w G^T)[i][l],  il = i*4+l
// ---------------------------------------------------------------------------
__global__ void wino_weight_transform(const float* __restrict__ w, float* __restrict__ U)
{
    int idx = blockIdx.x * blockDim.x + threadIdx.x;
    if (idx >= OCH * CIN) return;
    int o = idx >> 8;
    int c = idx & 255;
    const float* wp = w + (size_t)(o * CIN + c) * 9;

    float g[3][3];
#pragma unroll
    for (int r = 0; r < 3; ++r)
#pragma unroll
        for (int s = 0; s < 3; ++s) g[r][s] = wp[r * 3 + s];

    float u[4][3];
#pragma unroll
    for (int s = 0; s < 3; ++s) {
        u[0][s] = g[0][s];
        u[1][s] = 0.5f * (g[0][s] + g[1][s] + g[2][s]);
        u[2][s] = 0.5f * (g[0][s] - g[1][s] + g[2][s]);
        u[3][s] = g[2][s];
    }
#pragma unroll
    for (int r = 0; r < 4; ++r) {
        float a = u[r][0], b = u[r][1], d = u[r][2];
        float e[4];
        e[0] = a;
        e[1] = 0.5f * (a + b + d);
        e[2] = 0.5f * (a - b + d);
        e[3] = d;
#pragma unroll
        for (int l = 0; l < 4; ++l)
            U[(size_t)((r * 4 + l) * OCH + o) * CIN + c] = e[l];
    }
}

// ---------------------------------------------------------------------------
// Kernel 2: fused input-transform -> 16 GEMMs (f32 WMMA) -> output-transform
// grid = (TILES_TOTAL/TPB, OCH/OPB), block = 512 (16 waves, wave w owns il=w)
// ---------------------------------------------------------------------------
__launch_bounds__(512)
__global__ void wino_main(const float* __restrict__ x, const float* __restrict__ U,
                          float* __restrict__ out)
{
    // Phase 1 (GEMM): smem = V[il=16][tile=16][c stride VSTRIDE]  (34816 B)
    // Phase 2 (epilogue): smem = M[o=64][tile=16][il=16]          (65536 B)
    __shared__ float smem[OPB * 16 * 16];   // 16384 floats = 64 KB (covers both phases)

    const int tb    = blockIdx.x;          // tile block
    const int ob    = blockIdx.y;          // output-channel block
    const int tid   = threadIdx.x;
    const int lane  = tid & 31;
    const int il    = tid >> 5;            // wave id == transform position (0..15)
    const int obase = ob * OPB;

    // ---- per-thread (tile, channel-within-chunk) for the load/transform phase
    const int tile_l = tid & 15;           // 0..15
    const int cl     = tid >> 4;           // 0..31
    const int tile_g = tb * TPB + tile_l;
    const int n      = tile_g / TILES_PER_N;
    const int trem   = tile_g % TILES_PER_N;
    const int th     = trem / TW;
    const int tw     = trem % TW;
    const int ih0    = 2 * th - 1;
    const int iw0    = 2 * tw - 1;

    // Branchless boundary handling: clamp addresses, zero via 0/1 multiplier.
    int   rofs[4], cofs[4];
    float rmask[4], cmask[4];
#pragma unroll
    for (int r = 0; r < 4; ++r) {
        const int ih = ih0 + r;
        const bool rok = (unsigned)ih < (unsigned)H_DIM;
        rofs[r]  = (rok ? ih : 0) * W_DIM;
        rmask[r] = rok ? 1.f : 0.f;
    }
#pragma unroll
    for (int s = 0; s < 4; ++s) {
        const int iw = iw0 + s;
        const bool cok = (unsigned)iw < (unsigned)W_DIM;
        cofs[s]  = cok ? iw : 0;
        cmask[s] = cok ? 1.f : 0.f;
    }

    v8f acc[OFRAGS];
#pragma unroll
    for (int f = 0; f < OFRAGS; ++f)
        acc[f] = (v8f){0.f, 0.f, 0.f, 0.f, 0.f, 0.f, 0.f, 0.f};

    // WMMA fragment addressing (constant over the K loop)
    const int m16   = lane & 15;
    const int khalf = (lane >> 4) << 1;             // lanes 16..31 hold K+2,K+3
    const float* Arow[OFRAGS];
#pragma unroll
    for (int f = 0; f < OFRAGS; ++f)
        Arow[f] = U + ((size_t)il * OCH + (obase + f * 16 + m16)) * CIN;
    const float* Bbase = &smem[il * (16 * VSTRIDE) + m16 * VSTRIDE];

    for (int cb = 0; cb < CIN; cb += CCHUNK) {
        // ---- load one 4x4 input patch (unconditional, masked) ; V = BT d BT^T
        const int c = cb + cl;
        const float* xp = x + ((size_t)(n * CIN + c) * H_DIM) * W_DIM;
        float d[4][4];
#pragma unroll
        for (int r = 0; r < 4; ++r)
#pragma unroll
            for (int s = 0; s < 4; ++s)
                d[r][s] = xp[rofs[r] + cofs[s]] * (rmask[r] * cmask[s]);

        float t[4][4];
#pragma unroll
        for (int s = 0; s < 4; ++s) {                 // rows: BT * d
            t[0][s] = d[0][s] - d[2][s];
            t[1][s] = d[1][s] + d[2][s];
            t[2][s] = d[2][s] - d[1][s];
            t[3][s] = d[1][s] - d[3][s];
        }
#pragma unroll
        for (int r = 0; r < 4; ++r) {                 // cols: * BT^T
            float v0 = t[r][0] - t[r][2];
            float v1 = t[r][1] + t[r][2];
            float v2 = t[r][2] - t[r][1];
            float v3 = t[r][1] - t[r][3];
            smem[(r * 4 + 0) * (16 * VSTRIDE) + tile_l * VSTRIDE + cl] = v0;
            smem[(r * 4 + 1) * (16 * VSTRIDE) + tile_l * VSTRIDE + cl] = v1;
            smem[(r * 4 + 2) * (16 * VSTRIDE) + tile_l * VSTRIDE + cl] = v2;
            smem[(r * 4 + 3) * (16 * VSTRIDE) + tile_l * VSTRIDE + cl] = v3;
        }
        __syncthreads();

        // ---- GEMM over this K chunk: D(16x16) += A(16x4) * B(4x16) per step
#pragma unroll
        for (int k = 0; k < CCHUNK; k += 4) {
            const int kk = k + khalf;
            v2f b = *(const v2f*)&Bbase[kk];
#pragma unroll
            for (int f = 0; f < OFRAGS; ++f) {
                v2f a = *(const v2f*)&Arow[f][cb + kk];
                acc[f] = __builtin_amdgcn_wmma_f32_16x16x4_f32(
                    false, a, false, b, (short)0, acc[f], false, false);
            }
        }
        __syncthreads();   // smem reused next chunk
    }

    // ---- scatter accumulators to smem as M[o_local][tile][il]
    // C/D layout: vgpr r -> M = r (+8 if lane>=16), N = lane&15
    {
        const int orow = (lane >> 4) << 3;            // +8 for upper half-wave
        const int ncol = lane & 15;
#pragma unroll
        for (int f = 0; f < OFRAGS; ++f)
#pragma unroll
            for (int r = 0; r < 8; ++r)
                smem[(((f * 16 + r + orow) * 16) + ncol) * 16 + il] = acc[f][r];
    }
    __syncthreads();

    // ---- output transform: Y = AT M AT^T, (o_local, tile) pairs over threads
    for (int p = tid; p < OPB * 16; p += 512) {
        const int o_l = p >> 4;                       // 0..63
        const int tl  = p & 15;
        const int tg  = tb * TPB + tl;
        const int nn  = tg / TILES_PER_N;
        const int tr  = tg % TILES_PER_N;
        const int hh  = (tr / TW) * 2;
        const int ww  = (tr % TW) * 2;

        float m[16];
#pragma unroll
        for (int i = 0; i < 16; ++i)
            m[i] = smem[(o_l * 16 + tl) * 16 + i];

        float z0[4], z1[4];
#pragma unroll
        for (int c = 0; c < 4; ++c) {
            z0[c] = m[0 * 4 + c] + m[1 * 4 + c] + m[2 * 4 + c];
            z1[c] = m[1 * 4 + c] - m[2 * 4 + c] - m[3 * 4 + c];
        }
        const float y00 = z0[0] + z0[1] + z0[2];
        const float y01 = z0[1] - z0[2] - z0[3];
        const float y10 = z1[0] + z1[1] + z1[2];
        const float y11 = z1[1] - z1[2] - z1[3];

        const int o = obase + o_l;
        float* op = out + (((size_t)(nn * OCH + o) * H_DIM) + hh) * W_DIM + ww;
        op[0]         = y00;
        op[1]         = y01;
        op[W_DIM]     = y10;
        op[W_DIM + 1] = y11;
    }
}

// ---------------------------------------------------------------------------
extern "C" void kernel_launch(void* const* d_in, const int* in_sizes, int n_in,
                              void* d_out, int out_size, void* d_ws, size_t ws_size,
                              hipStream_t stream)
{
    const float* x = (const float*)d_in[0];   // (16,256,56,56) fp32
    const float* w = (const float*)d_in[1];   // (256,256,3,3) fp32
    float* out = (float*)d_out;               // (16,256,56,56) fp32
    float* U = (float*)d_ws;                  // 16*256*256 floats = 4 MB scratch

    wino_weight_transform<<<dim3((OCH * CIN + 255) / 256), dim3(256), 0, stream>>>(w, U);
    wino_main<<<dim3(TILES_TOTAL / TPB, OCH / OPB), dim3(512), 0, stream>>>(x, U, out);
}